// PtModule_76166950027779
// MI455X (gfx1250) — compile-verified
//
#include <hip/hip_runtime.h>

// y = ((x + 10) * 2)^2 on 8192x8192 fp32.
// Pure streaming: 512 MB of traffic @ 23.3 TB/s => ~22 us floor, ~0.2 GFLOP.
// Strategy: B128 vector accesses, non-temporal cache hints, and the CDNA5
// async global->LDS pipeline (GLOBAL_LOAD_ASYNC_TO_LDS_B128 + ASYNCcnt
// double-buffering) so each wave always has the next tile's DRAM fetch in
// flight while computing/storing the current one.

typedef float v4f __attribute__((ext_vector_type(4)));
// 32-bit LDS (addrspace(3)) pointer type for the async-load VDST operand.
typedef __attribute__((address_space(3))) v4f lds_v4f;

#define TPB 256  // 8 wave32 waves per block

__global__ __launch_bounds__(TPB) void fused_square_stream_kernel(
    const float* __restrict__ x, float* __restrict__ y, long long nvec4)
{
    // Double-buffered staging: 2 * 256 * 16B = 8 KB LDS per block.
    // Each thread owns one 16B slot per buffer; each wave only reads LDS it
    // wrote itself via its own async load, so no workgroup barrier is needed —
    // ordering is purely the per-wave ASYNCcnt.
    __shared__ v4f buf[2][TPB];

    const int tid = threadIdx.x;
    const long long stride = (long long)gridDim.x * TPB;
    long long idx = (long long)blockIdx.x * TPB + tid;

    const v4f* __restrict__ xv = (const v4f*)x;
    v4f* __restrict__ yv = (v4f*)y;

    // Raw 32-bit LDS addresses of this thread's two staging slots.
    lds_v4f* b0 = (lds_v4f*)&buf[0][tid];
    lds_v4f* b1 = (lds_v4f*)&buf[1][tid];

    // Prologue: kick off the async DRAM->LDS fetch of tile 0 (ASYNCcnt = 1).
    if (idx < nvec4) {
        asm volatile(
            "global_load_async_to_lds_b128 %0, %1, off th:TH_LOAD_NT"
            :: "v"(b0), "v"(xv + idx)
            : "memory");
    }

    int sel = 0;
    for (; idx < nvec4; idx += stride) {
        const long long nxt = idx + stride;
        if (nxt < nvec4) {
            // Issue tile i+1 into the other buffer (ASYNCcnt -> 2), then wait
            // until <=1 outstanding: tile i has landed in LDS, tile i+1 keeps
            // the memory system busy while we compute below.
            asm volatile(
                "global_load_async_to_lds_b128 %0, %1, off th:TH_LOAD_NT"
                :: "v"(sel ? b0 : b1), "v"(xv + nxt)
                : "memory");
            asm volatile("s_wait_asynccnt 0x1" ::: "memory");
        } else {
            // Last tile: drain.
            asm volatile("s_wait_asynccnt 0x0" ::: "memory");
        }

        // ds_load_b128 of the tile that just completed.
        v4f v = buf[sel][tid];

        // ((x + 10) * 2)^2
        v4f t = (v + 10.0f) * 2.0f;
        v4f r = t * t;

        // Streaming store: global_store_b128 with non-temporal hint so the
        // 256 MB output doesn't rinse through L2.
        __builtin_nontemporal_store(r, yv + idx);

        sel ^= 1;
    }
}

// Scalar tail for element counts not divisible by 4 (not hit for 8192^2,
// but keeps the kernel correct for any size).
__global__ void fused_square_tail_kernel(const float* __restrict__ x,
                                         float* __restrict__ y,
                                         long long start, long long n)
{
    long long i = start + (long long)blockIdx.x * blockDim.x + threadIdx.x;
    if (i < n) {
        float t = (x[i] + 10.0f) * 2.0f;
        y[i] = t * t;
    }
}

extern "C" void kernel_launch(void* const* d_in, const int* in_sizes, int n_in,
                              void* d_out, int out_size, void* d_ws, size_t ws_size,
                              hipStream_t stream)
{
    (void)n_in; (void)d_ws; (void)ws_size; (void)out_size;

    const float* x = (const float*)d_in[0];
    float* y = (float*)d_out;

    const long long n = (long long)in_sizes[0];   // 67,108,864 for 8192x8192
    const long long nvec4 = n >> 2;               // 16,777,216 float4 tiles

    // 8192 blocks x 256 threads = 2,097,152 threads -> exactly 8 pipeline
    // iterations per thread over 16M float4s; 65,536 wave32 waves in flight.
    long long want = (nvec4 + TPB - 1) / TPB;
    int blocks = (int)(want < 8192 ? (want > 0 ? want : 1) : 8192);

    fused_square_stream_kernel<<<blocks, TPB, 0, stream>>>(x, y, nvec4);

    const long long rem = n & 3LL;
    if (rem) {
        fused_square_tail_kernel<<<1, 64, 0, stream>>>(x, y, n - rem, n);
    }
}